// PhraseAveragePretrainedEmbedding_21079699489144
// MI455X (gfx1250) — compile-verified
//
#include <hip/hip_runtime.h>

// Problem constants (from the reference harness)
#define BB   8
#define TT   4095
#define SS   2048
#define DD   300
#define PAD  1
#define NDT  19            // ceil(300/16) d-tiles
#define DPAD (NDT * 16)    // 304: padded prefix row stride (floats)
#define PROWS (SS + 1)     // prefix rows per batch (row 0 is zero)

typedef __attribute__((ext_vector_type(2))) float v2f;
typedef __attribute__((ext_vector_type(8))) float v8f;

// ---------------------------------------------------------------------------
// Kernel 1: left-pack leaf tokens  (one 256-thread block per batch)
// is_leaf = (lo==hi) && (tok != PAD); leaves[b, cumsum-1] = tok; rest = PAD.
// ---------------------------------------------------------------------------
__global__ __launch_bounds__(256) void pack_leaves_kernel(
    const int* __restrict__ x, const int* __restrict__ idx,
    int* __restrict__ leaves) {
  const int b    = blockIdx.x;
  const int tid  = threadIdx.x;
  const int lane = tid & 31;
  const int wid  = tid >> 5;

  __shared__ int wsum[8];
  __shared__ int carry;

  for (int s = tid; s < SS; s += 256) leaves[b * SS + s] = PAD;
  if (tid == 0) carry = 0;
  __syncthreads();

  for (int base = 0; base < TT; base += 256) {
    const int t = base + tid;
    int isleaf = 0, tok = 0;
    if (t < TT) {
      const long long off = (long long)b * TT + t;
      const int lo = idx[2 * off + 0];
      const int hi = idx[2 * off + 1];
      tok    = x[off];
      isleaf = (lo == hi) && (tok != PAD);
    }
    const unsigned m    = (unsigned)__ballot(isleaf);       // wave32
    const int      excl = __popc(m & ((1u << lane) - 1u));  // lane-exclusive
    if (lane == 0) wsum[wid] = __popc(m);
    __syncthreads();
    int woff = 0;
    for (int w = 0; w < wid; ++w) woff += wsum[w];
    const int pos = carry + woff + excl;
    if (isleaf && pos < SS) leaves[b * SS + pos] = tok;
    __syncthreads();
    if (tid == 0) {
      int tot = 0;
      for (int w = 0; w < 8; ++w) tot += wsum[w];
      carry += tot;
    }
    __syncthreads();
  }
}

// ---------------------------------------------------------------------------
// Kernel 2: column-wise prefix sums of gathered embeddings via WMMA.
//   P[b, s+1, :] = P[b, s, :] + W[leaves[b,s], :]
// Blocked scan: per 16-row tile,  D = L16 x E_tile + broadcast(carry),
// L16 = lower-triangular ones, as 4 chained K=4 fp32 WMMAs.
// One wave per (batch, 16-wide d-tile); 8*19 = 152 waves, 128 tiles each.
// Branch-free inner loop (EXEC all-ones); gathers double-buffered so the
// next tile's index->embedding load chain overlaps the WMMA+carry chain.
// ---------------------------------------------------------------------------
__global__ __launch_bounds__(32) void prefix_scan_wmma_kernel(
    const int* __restrict__ leaves, const float* __restrict__ W,
    float* __restrict__ P) {
  const int b    = blockIdx.x / NDT;
  const int dt   = blockIdx.x % NDT;
  const int lane = threadIdx.x;        // 0..31
  const int n    = lane & 15;          // column within tile
  const int half = lane >> 4;          // half-wave
  const int dcol = dt * 16 + n;        // padded column index (< DPAD)
  // Clamp instead of mask: padded P columns (300..303) duplicate col 299 and
  // are never read by the output kernel -> loads stay unconditional.
  const int dc   = (dcol < DD) ? dcol : (DD - 1);

  float* Pb = P + (size_t)b * PROWS * DPAD;
  Pb[dcol] = 0.0f;                     // prefix row s=0 (both halves: same val)

  // A chunks: L16[:, 4k..4k+3].  A-matrix f32 16x4 layout:
  //   lanes 0-15: M=lane, VGPR0=K0, VGPR1=K1; lanes 16-31: K2, K3.
  v2f A[4];
#pragma unroll
  for (int k = 0; k < 4; ++k) {
    const int k0 = 4 * k + 2 * half;
    A[k].x = (k0     <= n) ? 1.0f : 0.0f;
    A[k].y = (k0 + 1 <= n) ? 1.0f : 0.0f;
  }

  const int* lv = leaves + b * SS;

  // B chunk loader: E[sb+4k .. sb+4k+3, :] (K x N = 4 x 16), N over lanes
  // 0-15; K rows {0,1} in lanes 0-15 / {2,3} in lanes 16-31 of VGPR0/1.
  auto load_tile = [&](int sb, v2f* Bt) {
#pragma unroll
    for (int k = 0; k < 4; ++k) {
      const int r0 = sb + 4 * k + 2 * half;           // even -> 8B aligned
      const int2 pr = *(const int2*)(lv + r0);        // one b64 load
      Bt[k].x = W[(size_t)pr.x * DD + dc];
      Bt[k].y = W[(size_t)pr.y * DD + dc];
    }
  };

  float carry = 0.0f;  // running column sum, replicated across half-waves
  v2f Bcur[4];
  load_tile(0, Bcur);

  for (int sb = 0; sb < SS; sb += 16) {
    // Prefetch next tile's gathers before the dependent WMMA chain.
    v2f Bnext[4];
    if (sb + 16 < SS) load_tile(sb + 16, Bnext);

    v8f d;
#pragma unroll
    for (int r = 0; r < 8; ++r) d[r] = carry;   // C = carry bcast to rows

#pragma unroll
    for (int k = 0; k < 4; ++k)
      d = __builtin_amdgcn_wmma_f32_16x16x4_f32(false, A[k], false, Bcur[k],
                                                (short)0, d, false, false);

    // Store tile: lane holds (M = r + 8*half, N = n) in d[r].
    float* tile = Pb + (size_t)(sb + 1) * DPAD + dcol;
#pragma unroll
    for (int r = 0; r < 8; ++r)
      tile[(size_t)(r + 8 * half) * DPAD] = d[r];

    // New carry = inclusive prefix row 15 = VGPR7 of lanes 16..31.
    carry = __shfl(d[7], 16 + n, 32);

#pragma unroll
    for (int k = 0; k < 4; ++k) Bcur[k] = Bnext[k];
  }
}

// ---------------------------------------------------------------------------
// Kernel 3: out[b,t,:] = (P[b,hi+1,:] - P[b,lo,:]) / max(hi-lo+1, 1)
// One 128-thread block per (b,t); fully coalesced.
// ---------------------------------------------------------------------------
__global__ __launch_bounds__(128) void phrase_avg_out_kernel(
    const int* __restrict__ idx, const float* __restrict__ P,
    float* __restrict__ out) {
  const int bt = blockIdx.x;                 // 0 .. B*T-1
  const int b  = bt / TT;
  const int lo = idx[2 * (long long)bt + 0];
  const int hi = idx[2 * (long long)bt + 1];
  int cnt = hi - lo + 1;
  if (cnt < 1) cnt = 1;
  const float inv = 1.0f / (float)cnt;
  const float* Plo = P + ((size_t)b * PROWS + lo)     * DPAD;
  const float* Phi = P + ((size_t)b * PROWS + hi + 1) * DPAD;
  float* o = out + (size_t)bt * DD;
  for (int d = threadIdx.x; d < DD; d += 128)
    o[d] = (Phi[d] - Plo[d]) * inv;
}

// ---------------------------------------------------------------------------
extern "C" void kernel_launch(void* const* d_in, const int* in_sizes, int n_in,
                              void* d_out, int out_size, void* d_ws,
                              size_t ws_size, hipStream_t stream) {
  const int*   x   = (const int*)d_in[0];      // [B,T] tokens
  const int*   idx = (const int*)d_in[1];      // [B,T,2] (lo,hi)
  const float* W   = (const float*)d_in[2];    // [V,D] fp32
  float*       out = (float*)d_out;            // [B,T,D] fp32

  // Workspace layout: leaves (B*S int, 64KB) | P (B*(S+1)*DPAD fp32, ~20MB)
  char* ws = (char*)d_ws;
  int*  leaves = (int*)ws;
  const size_t leaves_bytes =
      (((size_t)BB * SS * sizeof(int)) + 255u) & ~(size_t)255u;
  float* P = (float*)(ws + leaves_bytes);

  pack_leaves_kernel<<<BB, 256, 0, stream>>>(x, idx, leaves);
  prefix_scan_wmma_kernel<<<BB * NDT, 32, 0, stream>>>(leaves, W, P);
  phrase_avg_out_kernel<<<BB * TT, 128, 0, stream>>>(idx, P, out);
}